// MaxPooling3D_19370302505453
// MI455X (gfx1250) — compile-verified
//
#include <hip/hip_runtime.h>

// ---------------------------------------------------------------------------
// Fused: 2x2 spatial max-pool (stride 2) -> depth sliding max (win=3) ->
//        affine (y*k + b) -> LeakyReLU(0.1)
// x:      [8, 16, 112, 112, 32]  f32
// kernel: [14, 56, 56, 32]       f32
// bias:   [14, 56, 56, 32]       f32
// out:    [8, 14, 56, 56, 32]    f32
// Pure bandwidth problem (~262 MB total => ~11us at 23.3 TB/s). Strategy:
// read x exactly once with NT-hinted coalesced b128 streams, keep kernel/bias
// resident in L2 (reused across batch), NT-store the output.
// ---------------------------------------------------------------------------

typedef float v4f __attribute__((ext_vector_type(4)));

namespace {
constexpr int Bn  = 8;
constexpr int Dn  = 16;
constexpr int Hn  = 112;
constexpr int Wn  = 112;
constexpr int Cn  = 32;
constexpr int OHn = Hn / 2;           // 56
constexpr int OWn = Wn / 2;           // 56
constexpr int ODn = Dn - 3 + 1;       // 14
constexpr float kAlpha = 0.1f;
constexpr int kThreadsTotal = Bn * OHn * OWn * (Cn / 4);  // 200704
constexpr int kBlock = 256;
}  // namespace

__device__ __forceinline__ v4f vmax4(v4f a, v4f b) {
  v4f r;
  r.x = fmaxf(a.x, b.x);
  r.y = fmaxf(a.y, b.y);
  r.z = fmaxf(a.z, b.z);
  r.w = fmaxf(a.w, b.w);
  return r;
}

__device__ __forceinline__ v4f nt_load(const float* p) {
  return __builtin_nontemporal_load((const v4f*)p);
}

__global__ __launch_bounds__(kBlock) void MaxPooling3D_fused_kernel(
    const float* __restrict__ x, const float* __restrict__ krn,
    const float* __restrict__ bia, float* __restrict__ out) {
  const int tid = blockIdx.x * kBlock + threadIdx.x;
  if (tid >= kThreadsTotal) return;

  // Decompose: c4 fastest (8 threads cover one pixel's 32 channels = 128B).
  const int c = (tid & 7) * 4;
  int t = tid >> 3;
  const int ow = t % OWn;
  t /= OWn;
  const int oh = t % OHn;
  const int b  = t / OHn;

  const size_t strideH = (size_t)Wn * Cn;         // 3584 floats
  const size_t strideD = (size_t)Hn * strideH;    // 401408 floats
  const size_t planeO  = (size_t)OHn * OWn * Cn;  // 100352 floats

  const float* xb = x + (size_t)b * Dn * strideD
                      + (size_t)(oh * 2) * strideH
                      + (size_t)(ow * 2) * Cn + c;

  const size_t po = (size_t)(oh * OWn + ow) * Cn + c;
  const float* wb = krn + po;
  const float* bb = bia + po;
  float* ob       = out + (size_t)b * ODn * planeO + po;

  v4f p0 = {};  // pool[d-2]
  v4f p1 = {};  // pool[d-1]

#pragma unroll
  for (int d = 0; d < Dn; ++d) {
    const float* xp = xb + (size_t)d * strideD;

    // gfx1250 global_prefetch_b8: pull depth d+2 toward the caches while the
    // current depth's loads are in flight.
    if (d + 2 < Dn) {
      __builtin_prefetch(xp + 2 * strideD, 0, 1);
    }

    // 2x2 spatial pool: four NT-hinted b128 loads, each element read once.
    v4f a0 = nt_load(xp);
    v4f a1 = nt_load(xp + Cn);
    v4f a2 = nt_load(xp + strideH);
    v4f a3 = nt_load(xp + strideH + Cn);
    v4f pd = vmax4(vmax4(a0, a1), vmax4(a2, a3));

    if (d >= 2) {
      const int od = d - 2;
      v4f y  = vmax4(p0, vmax4(p1, pd));
      v4f kk = *(const v4f*)(wb + (size_t)od * planeO);  // RT: L2-resident
      v4f bv = *(const v4f*)(bb + (size_t)od * planeO);
      v4f z  = y * kk + bv;
      v4f r;
      r.x = (z.x >= 0.f) ? z.x : kAlpha * z.x;
      r.y = (z.y >= 0.f) ? z.y : kAlpha * z.y;
      r.z = (z.z >= 0.f) ? z.z : kAlpha * z.z;
      r.w = (z.w >= 0.f) ? z.w : kAlpha * z.w;
      __builtin_nontemporal_store(r, (v4f*)(ob + (size_t)od * planeO));
    }
    p0 = p1;
    p1 = pd;
  }
}

extern "C" void kernel_launch(void* const* d_in, const int* in_sizes, int n_in,
                              void* d_out, int out_size, void* d_ws, size_t ws_size,
                              hipStream_t stream) {
  (void)in_sizes; (void)n_in; (void)out_size; (void)d_ws; (void)ws_size;
  const float* x   = (const float*)d_in[0];
  const float* krn = (const float*)d_in[1];
  const float* bia = (const float*)d_in[2];
  float* out       = (float*)d_out;

  const int grid = (kThreadsTotal + kBlock - 1) / kBlock;  // 784
  MaxPooling3D_fused_kernel<<<grid, kBlock, 0, stream>>>(x, krn, bia, out);
}